// Aggregation_6081673691381
// MI455X (gfx1250) — compile-verified
//
#include <hip/hip_runtime.h>
#include <cstdint>

// Native CDNA5 f32 max atomic (IEEE maximumNumber semantics), no-return form,
// saddr addressing: 32-bit per-lane voffset + uniform 64-bit SGPR base.
// scope:SCOPE_DEV -> RMW performed at device scope (L2), coherent across WGPs.
#define ATOMIC_MAX_F32(VOFF, VAL, SBASE, OFS)                                  \
  asm volatile("global_atomic_max_num_f32 %0, %1, %2 offset:" #OFS             \
               " scope:SCOPE_DEV"                                              \
               :: "v"(VOFF), "v"(VAL), "s"(SBASE) : "memory")

#define FEAT 32  // D from the reference (x is (E, 32) float32)

__global__ void init_neginf_kernel(float4* __restrict__ out, int n4) {
  int i = blockIdx.x * blockDim.x + threadIdx.x;
  if (i < n4) {
    const float ninf = -__builtin_inff();
    out[i] = make_float4(ninf, ninf, ninf, ninf);
  }
}

// One thread per (edge, group-of-4-features). D=32 -> 8 groups per edge.
// A wave32 covers 4 consecutive edges = 512 contiguous bytes of x (b128 loads).
__global__ void scatter_max_kernel(const float4* __restrict__ x4,
                                   const int64_t* __restrict__ tgt,
                                   float* __restrict__ out,
                                   long long n_items) {
  long long t = (long long)blockIdx.x * blockDim.x + threadIdx.x;
  if (t >= n_items) return;

  long long e = t >> 3;          // edge index (D/4 == 8 groups per edge)
  unsigned g  = (unsigned)(t & 7);

  float4 v = x4[t];              // == x[e*32 + g*4 .. +3], coalesced b128

  long long node = tgt[e];       // int64 targets; lane-uniform within 8 lanes
  unsigned voff = (unsigned)node * (FEAT * 4u) + g * 16u;  // byte offset into out

  ATOMIC_MAX_F32(voff, v.x, out, 0);
  ATOMIC_MAX_F32(voff, v.y, out, 4);
  ATOMIC_MAX_F32(voff, v.z, out, 8);
  ATOMIC_MAX_F32(voff, v.w, out, 12);
}

// Inputs are finite, so a surviving -inf means "empty segment" -> 0 (matches
// the reference's where(isneginf(agg), 0, agg)).
__global__ void finalize_kernel(float4* __restrict__ out, int n4) {
  int i = blockIdx.x * blockDim.x + threadIdx.x;
  if (i >= n4) return;
  float4 v = out[i];
  const float ninf = -__builtin_inff();
  v.x = (v.x == ninf) ? 0.0f : v.x;
  v.y = (v.y == ninf) ? 0.0f : v.y;
  v.z = (v.z == ninf) ? 0.0f : v.z;
  v.w = (v.w == ninf) ? 0.0f : v.w;
  out[i] = v;
}

extern "C" void kernel_launch(void* const* d_in, const int* in_sizes, int n_in,
                              void* d_out, int out_size, void* d_ws, size_t ws_size,
                              hipStream_t stream) {
  const float*   x  = (const float*)d_in[0];          // (E, 32) f32
  const int64_t* ei = (const int64_t*)d_in[1];        // (2, E) i64
  (void)d_ws; (void)ws_size; (void)n_in;

  const long long E   = (long long)in_sizes[1] / 2;
  const int64_t*  tgt = ei + E;                       // row 1 = targets
  float*          out = (float*)d_out;                // (N, 32) f32
  const int       n4  = out_size / 4;                 // N*32/4 float4's

  const int TB = 256;

  // 1) out = -inf
  init_neginf_kernel<<<(n4 + TB - 1) / TB, TB, 0, stream>>>((float4*)out, n4);

  // 2) scatter-max with native f32 NUM-max atomics (L2-resident output)
  const long long items  = E * (FEAT / 4);            // E * 8
  const long long blocks = (items + TB - 1) / TB;
  scatter_max_kernel<<<dim3((unsigned)blocks), TB, 0, stream>>>(
      (const float4*)x, tgt, out, items);

  // 3) -inf (empty segment) -> 0
  finalize_kernel<<<(n4 + TB - 1) / TB, TB, 0, stream>>>((float4*)out, n4);
}